// MultiHeadSelfAttention_36704790512351
// MI455X (gfx1250) — compile-verified
//
#include <hip/hip_runtime.h>
#include <hip/hip_bf16.h>
#include <math.h>

// ---------------------------------------------------------------------------
// MHA on gfx1250 (MI455X), fp32 end-to-end via V_WMMA_F32_16X16X4_F32.
//   Stage 1: qkv = x @ W_qkv + b_qkv            (4096x3072, K=1024)
//   Stage 2: flash attention per (b,h), 128 query rows / workgroup
//            K tiles DMA'd by the Tensor Data Mover (tensor_load_to_lds)
//   Stage 3: out = attn @ W_out + b_out         (4096x1024, K=1024)
// B/V matrices are stored k-pair interleaved in LDS so every WMMA B-fragment
// is a single contiguous ds_load_b64 (no VGPR packing movs).
// ---------------------------------------------------------------------------

typedef float v2f __attribute__((ext_vector_type(2)));
typedef float v8f __attribute__((ext_vector_type(8)));
typedef unsigned int u32x4 __attribute__((ext_vector_type(4)));
typedef int i32x4 __attribute__((ext_vector_type(4)));
typedef int i32x8 __attribute__((ext_vector_type(8)));

#if defined(__HIP_DEVICE_COMPILE__) && \
    !__has_builtin(__builtin_amdgcn_wmma_f32_16x16x4_f32)
#error "missing __builtin_amdgcn_wmma_f32_16x16x4_f32 on device toolchain"
#endif

#if defined(__HIP_DEVICE_COMPILE__) && \
    __has_builtin(__builtin_amdgcn_tensor_load_to_lds)
#define USE_TDM 1
#else
#define USE_TDM 0
#endif

__device__ __forceinline__ v8f wmma_f32(v2f a, v2f b, v8f c) {
#if defined(__HIP_DEVICE_COMPILE__)
  // (neg_a, A, neg_b, B, c_mod, C, reuse_a, reuse_b)
  return __builtin_amdgcn_wmma_f32_16x16x4_f32(false, a, false, b, (short)0, c,
                                               false, false);
#else
  (void)a; (void)b;
  return c;  // host pass never executes device code
#endif
}

// ---------------------------------------------------------------------------
// Generic GEMM + bias: C[M,N] = A[M,K] @ B[K,N] + bias[N]
// Block tile 64x128, K-chunk 32, 256 threads (8 waves, 4x2 wave grid),
// each wave produces a 16x64 strip as four 16x16 WMMA accumulators.
// ---------------------------------------------------------------------------
#define AS_STRIDE 36   // A: 64 rows x 32 k, padded; conflict-free a-frag b64s
#define BP_STRIDE 288  // B: per k-pair row (256 data + pad), 288%64==32 ->
                       // lane halves (k vs k+2) hit disjoint bank sets

__global__ __launch_bounds__(256) void gemm_bias_kernel(
    const float* __restrict__ A, const float* __restrict__ B,
    const float* __restrict__ bias, float* __restrict__ C,
    int M, int N, int K) {
  __shared__ __align__(16) float As[64 * AS_STRIDE];   //  9216 B
  __shared__ __align__(16) float Bs[16 * BP_STRIDE];   // 18432 B

  const int tid  = threadIdx.x;
  const int lane = tid & 31;
  const int wid  = tid >> 5;
  const int half = lane >> 4;   // 0: lanes 0-15, 1: lanes 16-31
  const int l16  = lane & 15;
  const int wm   = wid & 3;     // wave row (0..3) -> 16-row slabs
  const int wn   = wid >> 2;    // wave col (0..1) -> 64-col slabs

  const int blockRow = blockIdx.y * 64;
  const int blockCol = blockIdx.x * 128;

  v8f acc[4];
#pragma unroll
  for (int t = 0; t < 4; ++t) acc[t] = (v8f){0, 0, 0, 0, 0, 0, 0, 0};

  // staging registers (double buffer against global latency)
  float4 ra[2], rb0[2], rb1[2];

#pragma unroll
  for (int it = 0; it < 2; ++it) {
    int j = tid + 256 * it;
    ra[it] = *(const float4*)(A + (size_t)(blockRow + (j >> 3)) * K + (j & 7) * 4);
    int kp = j >> 5, c4 = j & 31;
    rb0[it] = *(const float4*)(B + (size_t)(2 * kp) * N + blockCol + c4 * 4);
    rb1[it] = *(const float4*)(B + (size_t)(2 * kp + 1) * N + blockCol + c4 * 4);
  }

  for (int k0 = 0; k0 < K; k0 += 32) {
    __syncthreads();
#pragma unroll
    for (int it = 0; it < 2; ++it) {
      int j = tid + 256 * it;
      *(float4*)(&As[(j >> 3) * AS_STRIDE + (j & 7) * 4]) = ra[it];
      int kp = j >> 5, c4 = j & 31;
#pragma unroll
      for (int e = 0; e < 4; ++e) {
        v2f pr = {(&rb0[it].x)[e], (&rb1[it].x)[e]};
        *(v2f*)(&Bs[kp * BP_STRIDE + (c4 * 4 + e) * 2]) = pr;
      }
    }
    __syncthreads();

    // prefetch next chunk into registers while computing this one
    if (k0 + 32 < K) {
      int kn = k0 + 32;
#pragma unroll
      for (int it = 0; it < 2; ++it) {
        int j = tid + 256 * it;
        ra[it] =
            *(const float4*)(A + (size_t)(blockRow + (j >> 3)) * K + kn + (j & 7) * 4);
        int kp = j >> 5, c4 = j & 31;
        rb0[it] =
            *(const float4*)(B + (size_t)(kn + 2 * kp) * N + blockCol + c4 * 4);
        rb1[it] =
            *(const float4*)(B + (size_t)(kn + 2 * kp + 1) * N + blockCol + c4 * 4);
      }
    }

#pragma unroll
    for (int kk = 0; kk < 32; kk += 4) {
      v2f a = *(const v2f*)(&As[(wm * 16 + l16) * AS_STRIDE + kk + 2 * half]);
      const int kpr = (kk >> 1) + half;  // (kk + 2*half)/2
#pragma unroll
      for (int t = 0; t < 4; ++t) {
        int n = wn * 64 + t * 16 + l16;
        v2f bfr = *(const v2f*)(&Bs[kpr * BP_STRIDE + n * 2]);
        acc[t] = wmma_f32(a, bfr, acc[t]);
      }
    }
  }

#pragma unroll
  for (int t = 0; t < 4; ++t) {
    int n    = blockCol + wn * 64 + t * 16 + l16;
    float bv = bias[n];
#pragma unroll
    for (int i = 0; i < 8; ++i) {
      int r = blockRow + wm * 16 + i + 8 * half;
      C[(size_t)r * N + n] = acc[t][i] + bv;
    }
  }
}

// ---------------------------------------------------------------------------
// TDM helpers: DMA a 32x64-f32 tile (row stride 3072 f32) into LDS with
// 4-DWORD padding after every 64 DWORDs -> LDS row stride 68 f32.
// ---------------------------------------------------------------------------
#define KL_STRIDE 68   // K row-major (68 = 64 + TDM pad): contiguous b64 frags
#define VP_STRIDE 160  // V k-pair interleaved: 160%64==32 -> conflict-free
#define PL_STRIDE 18

__device__ __forceinline__ unsigned lds_off(const void* p) {
  return (unsigned)(unsigned long long)(size_t)p;  // low 32b = LDS byte addr
}

__device__ __forceinline__ void wait_tensorcnt0() {
#if defined(__HIP_DEVICE_COMPILE__)
#if __has_builtin(__builtin_amdgcn_s_wait_tensorcnt)
  __builtin_amdgcn_s_wait_tensorcnt(0);
#else
  asm volatile("s_wait_tensorcnt 0x0" ::: "memory");
#endif
#endif
}
__device__ __forceinline__ void wait_tensorcnt1() {
#if defined(__HIP_DEVICE_COMPILE__)
#if __has_builtin(__builtin_amdgcn_s_wait_tensorcnt)
  __builtin_amdgcn_s_wait_tensorcnt(1);
#else
  asm volatile("s_wait_tensorcnt 0x1" ::: "memory");
#endif
#endif
}

#if USE_TDM
__device__ __forceinline__ void tdm_load_k_tile(const float* gsrc,
                                                unsigned ldsByteAddr) {
  unsigned long long ga = (unsigned long long)(size_t)gsrc;
  u32x4 g0;
  g0.x = 1u;                                   // count=1 (valid), user mode
  g0.y = ldsByteAddr;                          // D#.lds_addr
  g0.z = (unsigned)(ga & 0xffffffffu);         // global_addr[31:0]
  g0.w = (unsigned)((ga >> 32) & 0x01ffffffu)  // global_addr[56:32]
       | (2u << 30);                           // type = 2 ("image")
  i32x8 g1;
  g1[0] = (2 << 16)    // data_size = 4 bytes
        | (1 << 20)    // pad_enable
        | (5 << 22)    // pad_interval: 64 DWORDs
        | (3 << 25);   // pad_amount:   4 DWORDs  -> LDS row stride 68 f32
  g1[1] = 64 << 16;    // tensor_dim0[15:0] = 64 (bits 63:48)
  g1[2] = 32 << 16;    // tensor_dim0 hi = 0 | tensor_dim1[15:0] = 32
  g1[3] = 64 << 16;    // tensor_dim1 hi = 0 | tile_dim0 = 64
  g1[4] = 32;          // tile_dim1 = 32 | tile_dim2 = 0
  g1[5] = 3072;        // tensor_dim0_stride[31:0] = 3072 elements
  g1[6] = 0;           // stride hi | tensor_dim1_stride lo
  g1[7] = 0;
  i32x4 z4 = {0, 0, 0, 0};
#if __clang_major__ >= 23
  i32x8 z8 = {0, 0, 0, 0, 0, 0, 0, 0};
  __builtin_amdgcn_tensor_load_to_lds(g0, g1, z4, z4, z8, 0);
#else
  __builtin_amdgcn_tensor_load_to_lds(g0, g1, z4, z4, 0);
#endif
}
#endif

// ---------------------------------------------------------------------------
// Flash attention. qkv: [B*T][3072]   (cols: Q=h*64.., K=1024+h*64.., V=2048+)
// out: flat (B,H,T,D) order -> reading it later as (4096x1024) row-major
// reproduces the reference's no-transpose reshape exactly.
// Grid: x = T/128 = 16 (query blocks), y = B*H = 32.
// ---------------------------------------------------------------------------
__global__ __launch_bounds__(256) void attn_kernel(
    const float* __restrict__ qkv, float* __restrict__ out) {
  __shared__ __align__(16) float Kl[2][32 * KL_STRIDE];    // 2 x 8704 B
  __shared__ __align__(16) float Vl[16 * VP_STRIDE];       // 10240 B
  __shared__ __align__(16) float Pl[8 * 16 * PL_STRIDE];   //  9216 B

  const int tid  = threadIdx.x;
  const int lane = tid & 31;
  const int wid  = tid >> 5;
  const int half = lane >> 4;
  const int l16  = lane & 15;

  const int bh = blockIdx.y;   // b*16 + h
  const int b  = bh >> 4;
  const int h  = bh & 15;
  const int qbase = blockIdx.x * 128 + wid * 16;  // first query row (in T)

  // staging assignment: key-pair skp (0..15) x float4 column sc4 (0..15)
  const int skp = tid >> 4;
  const int sc4 = tid & 15;

  const float* kbase = qkv + (size_t)(b * 2048) * 3072 + h * 64 + 1024;

  // Load Q as 16 A-fragments (16 rows x 64 d), kept in VGPRs for the run.
  v2f qf[16];
  {
    const float* qp =
        qkv + (size_t)(b * 2048 + qbase + l16) * 3072 + h * 64 + 2 * half;
#pragma unroll
    for (int s = 0; s < 16; ++s) qf[s] = *(const v2f*)(qp + s * 4);
  }

  v8f acc[4];
#pragma unroll
  for (int t = 0; t < 4; ++t) acc[t] = (v8f){0, 0, 0, 0, 0, 0, 0, 0};
  float mrow[8], lrow[8];
#pragma unroll
  for (int i = 0; i < 8; ++i) {
    mrow[i] = -INFINITY;
    lrow[i] = 0.0f;
  }

  float* Pw = &Pl[wid * 16 * PL_STRIDE];  // per-wave staging, no x-wave sync

  // prologue: start K tile 0 (TDM) and stage first V tile into registers
#if USE_TDM
  if (wid == 0) tdm_load_k_tile(kbase, lds_off(&Kl[0][0]));
  float4 rv0, rv1;
#else
  float4 rk0, rk1, rv0, rv1;
#endif
  {
    const float* s0 = qkv + (size_t)(b * 2048 + 2 * skp) * 3072 + h * 64;
#if !USE_TDM
    rk0 = *(const float4*)(s0 + 1024 + sc4 * 4);
    rk1 = *(const float4*)(s0 + 3072 + 1024 + sc4 * 4);
#endif
    rv0 = *(const float4*)(s0 + 2048 + sc4 * 4);
    rv1 = *(const float4*)(s0 + 3072 + 2048 + sc4 * 4);
  }

  int ibuf = 0;
  for (int kb = 0; kb < 2048; kb += 32, ibuf ^= 1) {
    __syncthreads();  // everyone done with previous tile's compute
    // V rows go in k-pair interleaved
#pragma unroll
    for (int e = 0; e < 4; ++e) {
      v2f pr = {(&rv0.x)[e], (&rv1.x)[e]};
      *(v2f*)(&Vl[skp * VP_STRIDE + (sc4 * 4 + e) * 2]) = pr;
    }
#if USE_TDM
    // issue next K tile's DMA, then ensure the current one has landed
    if (wid == 0) {
      if (kb + 32 < 2048) {
        tdm_load_k_tile(kbase + (size_t)(kb + 32) * 3072,
                        lds_off(&Kl[ibuf ^ 1][0]));
        wait_tensorcnt1();  // <=1 outstanding -> current tile complete
      } else {
        wait_tensorcnt0();
      }
    }
#else
    *(float4*)(&Kl[ibuf][(2 * skp) * KL_STRIDE + sc4 * 4])     = rk0;
    *(float4*)(&Kl[ibuf][(2 * skp + 1) * KL_STRIDE + sc4 * 4]) = rk1;
#endif
    __syncthreads();  // publish K (TDM) + V (stores) to all waves

    // prefetch next V tile into registers
    if (kb + 32 < 2048) {
      const float* s0 =
          qkv + (size_t)(b * 2048 + kb + 32 + 2 * skp) * 3072 + h * 64;
#if !USE_TDM
      rk0 = *(const float4*)(s0 + 1024 + sc4 * 4);
      rk1 = *(const float4*)(s0 + 3072 + 1024 + sc4 * 4);
#endif
      rv0 = *(const float4*)(s0 + 2048 + sc4 * 4);
      rv1 = *(const float4*)(s0 + 3072 + 2048 + sc4 * 4);
    }

#pragma unroll
    for (int sub = 0; sub < 2; ++sub) {
      const float* Ks = &Kl[ibuf][sub * 16 * KL_STRIDE];

      // S = Q @ K^T  (16 queries x 16 keys, contraction over d=64)
      v8f s = (v8f){0, 0, 0, 0, 0, 0, 0, 0};
#pragma unroll
      for (int kk = 0; kk < 64; kk += 4) {
        v2f bfr = *(const v2f*)(&Ks[l16 * KL_STRIDE + kk + 2 * half]);
        s = wmma_f32(qf[kk >> 2], bfr, s);
      }

      // online softmax; row r = i + 8*half lives on the lanes of this half
      float sc[8];
#pragma unroll
      for (int i = 0; i < 8; ++i) {
        float si = s[i] * 0.125f;  // 1/sqrt(64)
        float mx = si;
#pragma unroll
        for (int msk = 8; msk; msk >>= 1)
          mx = fmaxf(mx, __shfl_xor(mx, msk, 32));
        float mn = fmaxf(mrow[i], mx);
        float pi = __expf(si - mn);
        float rs = pi;
#pragma unroll
        for (int msk = 8; msk; msk >>= 1) rs += __shfl_xor(rs, msk, 32);
        float corr = __expf(mrow[i] - mn);
        lrow[i] = lrow[i] * corr + rs;
        mrow[i] = mn;
        sc[i]   = corr;
        Pw[(i + 8 * half) * PL_STRIDE + l16] = pi;  // C-layout -> LDS
      }
#pragma unroll
      for (int t = 0; t < 4; ++t)
#pragma unroll
        for (int i = 0; i < 8; ++i) acc[t][i] *= sc[i];

      // intra-wave LDS RAW: P just stored, about to re-read in A-layout
      asm volatile("s_wait_dscnt 0" ::: "memory");

      // O += P @ V  (contraction over the 16 keys of this subtile)
#pragma unroll
      for (int ko = 0; ko < 16; ko += 4) {
        v2f a = *(const v2f*)(&Pw[l16 * PL_STRIDE + ko + 2 * half]);
        const int kpr = sub * 8 + (ko >> 1) + half;  // global k-pair row
#pragma unroll
        for (int t = 0; t < 4; ++t) {
          v2f bfr = *(const v2f*)(&Vl[kpr * VP_STRIDE + (t * 16 + l16) * 2]);
          acc[t] = wmma_f32(a, bfr, acc[t]);
        }
      }
    }
  }

  // normalize and store in (b,h,t,d) flat order
#pragma unroll
  for (int i = 0; i < 8; ++i) {
    float inv = 1.0f / lrow[i];
    int trow  = qbase + i + 8 * half;
    float* op = out + ((size_t)bh * 2048 + trow) * 64;
#pragma unroll
    for (int t = 0; t < 4; ++t) op[t * 16 + l16] = acc[t][i] * inv;
  }
}

// ---------------------------------------------------------------------------
extern "C" void kernel_launch(void* const* d_in, const int* in_sizes, int n_in,
                              void* d_out, int out_size, void* d_ws,
                              size_t ws_size, hipStream_t stream) {
  (void)in_sizes; (void)n_in; (void)out_size; (void)ws_size;
  const float* x     = (const float*)d_in[0];
  const float* W_qkv = (const float*)d_in[1];
  const float* b_qkv = (const float*)d_in[2];
  const float* W_out = (const float*)d_in[3];
  const float* b_out = (const float*)d_in[4];
  float* out = (float*)d_out;

  float* qkv  = (float*)d_ws;                    // 4096*3072 f32 = 48 MB
  float* attn = qkv + (size_t)4096 * 3072;       // 4096*1024 f32 = 16 MB

  // Stage 1: qkv = x @ W_qkv + b_qkv
  dim3 g1(3072 / 128, 4096 / 64);
  gemm_bias_kernel<<<g1, 256, 0, stream>>>(x, W_qkv, b_qkv, qkv, 4096, 3072,
                                           1024);
  // Stage 2: attention
  dim3 g2(2048 / 128, 32);
  attn_kernel<<<g2, 256, 0, stream>>>(qkv, attn);
  // Stage 3: out = attn @ W_out + b_out (attn read as 4096x1024 row-major
  // reproduces the reference's no-transpose reshape)
  dim3 g3(1024 / 128, 4096 / 64);
  gemm_bias_kernel<<<g3, 256, 0, stream>>>(attn, W_out, b_out, out, 4096, 1024,
                                           1024);
}